// NewsvendorLayer_32306744000825
// MI455X (gfx1250) — compile-verified
//
#include <hip/hip_runtime.h>
#include <math.h>

// Problem constants (match reference)
#define CF_F     0.7f
#define GAMMA_F  0.1f
#define N_SUPP   2048
#define N_BISECT 60
#define BLOCK    256
#define EPT      (N_SUPP / BLOCK)   // 8 elements per thread
#define NCHUNK   (N_SUPP / 4)       // 512 float4 chunks per row

typedef float v4f __attribute__((ext_vector_type(4)));  // native vec4 float

// ---- CDNA5 feature detection (device pass only) ----
#if defined(__HIP_DEVICE_COMPILE__)
  #if __has_builtin(__builtin_amdgcn_global_load_async_to_lds_b128) && \
      __has_builtin(__builtin_amdgcn_s_wait_asynccnt)
    #define USE_ASYNC_LDS 1
  #else
    #define USE_ASYNC_LDS 0
  #endif
  #if __has_builtin(__builtin_amdgcn_ballot_w32)
    #define BALLOT32(p) __builtin_amdgcn_ballot_w32(p)
  #else
    #define BALLOT32(p) ((unsigned)__ballot(p))
  #endif
#else
  #define USE_ASYNC_LDS 0
  #define BALLOT32(p) 0u
#endif

#if USE_ASYNC_LDS
typedef int v4i_t __attribute__((ext_vector_type(4)));
typedef __attribute__((address_space(1))) v4i_t* gv4_t;  // global int4*
typedef __attribute__((address_space(3))) v4i_t* lv4_t;  // LDS int4*
#endif

// Packed 2-value fp64 block reduction (one tree, two accumulators)
__device__ __forceinline__ void block_reduce2(double v0, double v1,
                                              double* red, int t,
                                              double* r0, double* r1) {
  red[t] = v0;
  red[BLOCK + t] = v1;
  __syncthreads();
  for (int off = BLOCK / 2; off > 0; off >>= 1) {
    if (t < off) {
      red[t]         += red[t + off];
      red[BLOCK + t] += red[BLOCK + t + off];
    }
    __syncthreads();
  }
  *r0 = red[0];
  *r1 = red[BLOCK];
  __syncthreads();
}

__global__ __launch_bounds__(BLOCK, 8)
void newsvendor_fused(const float* __restrict__ x,
                      const float* __restrict__ y,
                      float* __restrict__ out,
                      int nbatch) {
  __shared__ __align__(16) float  s_p[N_SUPP];   // row of x (weights)
  __shared__ __align__(16) float  s_y[N_SUPP];   // sorted support
  __shared__ double s_pre[N_SUPP];               // inclusive prefix of p (fp64)
  __shared__ double s_red[2 * BLOCK];            // scan / reduction scratch
  __shared__ float  s_z;

  const int t   = threadIdx.x;
  const int row = blockIdx.x;
  const float* __restrict__ xrow = x + (size_t)row * N_SUPP;

  // ---- Phase 0: stage x-row and y into LDS (async global->LDS, gfx1250) ----
#if USE_ASYNC_LDS
  {
    // 512 x b128 chunks per array, 2 per thread, coalesced
    __builtin_amdgcn_global_load_async_to_lds_b128(
        (gv4_t)(xrow + 4 * t),           (lv4_t)&s_p[4 * t],           0, 0);
    __builtin_amdgcn_global_load_async_to_lds_b128(
        (gv4_t)(xrow + 4 * (t + BLOCK)), (lv4_t)&s_p[4 * (t + BLOCK)], 0, 0);
    __builtin_amdgcn_global_load_async_to_lds_b128(
        (gv4_t)(y + 4 * t),              (lv4_t)&s_y[4 * t],           0, 0);
    __builtin_amdgcn_global_load_async_to_lds_b128(
        (gv4_t)(y + 4 * (t + BLOCK)),    (lv4_t)&s_y[4 * (t + BLOCK)], 0, 0);
    __builtin_amdgcn_s_wait_asynccnt(0);
  }
#else
  for (int j = 0; j < EPT; ++j) {
    int i = t + j * BLOCK;
    s_p[i] = xrow[i];
    s_y[i] = y[i];
  }
#endif
  __syncthreads();

  // ---- Phase 1: moments (fp64) + local prefix over contiguous segment ----
  double sp = 0.0, sq = 0.0, sqy = 0.0, sqy2 = 0.0, run = 0.0;
  const int base = t * EPT;
  for (int j = 0; j < EPT; ++j) {
    double p  = (double)s_p[base + j];
    double yy = (double)s_y[base + j];
    double q  = p * p;
    sp += p; sq += q; sqy += q * yy; sqy2 += q * yy * yy;
    run += p;
    s_pre[base + j] = run;              // thread-local inclusive prefix
  }

  // Hillis-Steele scan of per-thread totals (256 values)
  s_red[t] = run;
  __syncthreads();
  for (int off = 1; off < BLOCK; off <<= 1) {
    double add = (t >= off) ? s_red[t - off] : 0.0;
    __syncthreads();
    s_red[t] += add;
    __syncthreads();
  }
  double excl = s_red[t] - run;         // exclusive offset for this thread
  __syncthreads();                      // s_red about to be reused
  for (int j = 0; j < EPT; ++j) s_pre[base + j] += excl;

  double S_tot, Bm, A, C;
  block_reduce2(sp,  sq,   s_red, t, &S_tot, &Bm);  // sum p, sum p^2
  block_reduce2(sqy, sqy2, s_red, t, &A, &C);       // sum p^2 y, sum p^2 y^2
  // (block_reduce2 barriers also publish s_pre to wave 0)

  // ---- Phase 2: bisection on wave 0; ballot-based searchsorted in sorted y ----
  if (t < 32) {
    const double CFd  = (double)CF_F;
    const double Gd   = (double)GAMMA_F;
    const double EPSd = (double)1e-12f;
    float lo = 0.0f, hi = 1.0f;
    for (int it = 0; it < N_BISECT; ++it) {
      const float mid = 0.5f * (lo + hi);
      // k = #{ y_i <= mid }: 32-ary level 1 (chunk-of-64 maxima), then pairs
      unsigned b1 = BALLOT32(s_y[t * 64 + 63] <= mid);
      int base1 = __popc(b1) * 64;                        // complete chunks
      bool c2 = (base1 < N_SUPP) && (s_y[base1 + 2 * t + 1] <= mid);
      unsigned b2 = BALLOT32(c2);
      int k = base1 + 2 * __popc(b2);                     // complete pairs
      if (k < N_SUPP && s_y[k] <= mid) ++k;               // straggler
      const double S_le = (k > 0) ? s_pre[k - 1] : 0.0;

      const double md  = (double)mid;
      const double pin = S_le - CFd * S_tot;              // sum p * pinball'
      const double num = A - md * Bm;                     // sum p^2 e
      const double s2  = C - 2.0 * md * A + md * md * Bm + EPSd;  // ||p*e||^2
      // sign(pin - G*num/nrm) == sign(pin*nrm - G*num) since nrm > 0
      const double f = pin * sqrt(s2) - Gd * num;
      if (f < 0.0) lo = mid; else hi = mid;
    }
    if (t == 0) s_z = 0.5f * (lo + hi);
  }
  __syncthreads();
  const float z = s_z;

  // ---- Phase 3: outputs, vectorized b128 non-temporal stores ----
  const size_t BN = (size_t)nbatch * N_SUPP;
  float* __restrict__ z_out    = out;
  float* __restrict__ pin_out  = out + nbatch;
  float* __restrict__ err_out  = pin_out + BN;
  float* __restrict__ werr_out = err_out + BN;
  const size_t ro = (size_t)row * N_SUPP;

  for (int c = t; c < NCHUNK; c += BLOCK) {        // c = t, t + 256
    const v4f y4 = *(const v4f*)&s_y[4 * c];
    const v4f p4 = *(const v4f*)&s_p[4 * c];
    const v4f e4  = y4 - z;                         // broadcast subtract
    v4f pb4, w4;
    pb4.x = fmaxf(CF_F * e4.x, (CF_F - 1.0f) * e4.x);
    pb4.y = fmaxf(CF_F * e4.y, (CF_F - 1.0f) * e4.y);
    pb4.z = fmaxf(CF_F * e4.z, (CF_F - 1.0f) * e4.z);
    pb4.w = fmaxf(CF_F * e4.w, (CF_F - 1.0f) * e4.w);
    w4 = p4 * e4;
    __builtin_nontemporal_store(e4,  (v4f*)&err_out[ro + 4 * c]);
    __builtin_nontemporal_store(pb4, (v4f*)&pin_out[ro + 4 * c]);
    __builtin_nontemporal_store(w4,  (v4f*)&werr_out[ro + 4 * c]);
  }
  if (t == 0) z_out[row] = z;
}

extern "C" void kernel_launch(void* const* d_in, const int* in_sizes, int n_in,
                              void* d_out, int out_size, void* d_ws, size_t ws_size,
                              hipStream_t stream) {
  (void)n_in; (void)out_size; (void)d_ws; (void)ws_size;
  const float* x = (const float*)d_in[0];
  const float* y = (const float*)d_in[1];
  float* out = (float*)d_out;
  const int n = in_sizes[1];           // 2048 support points
  const int b = in_sizes[0] / n;       // 4096 batch rows
  hipLaunchKernelGGL(newsvendor_fused, dim3(b), dim3(BLOCK), 0, stream,
                     x, y, out, b);
}